// TrafficGNN_50895362458313
// MI455X (gfx1250) — compile-verified
//
#include <hip/hip_runtime.h>
#include <hip/hip_bf16.h>

typedef __attribute__((ext_vector_type(2))) float v2f;
typedef __attribute__((ext_vector_type(8))) float v8f;

// ---------------------------------------------------------------------------
// init: deg = 1.0 (self loop), acc1 = 0, acc2 = 0
// ---------------------------------------------------------------------------
__global__ void gcn_init(float* __restrict__ deg, float* __restrict__ acc1,
                         float* __restrict__ acc2, int N) {
  long long i = (long long)blockIdx.x * blockDim.x + threadIdx.x;
  long long stride = (long long)gridDim.x * blockDim.x;
  for (long long j = i; j < (long long)N; j += stride) deg[j] = 1.0f;
  for (long long j = i; j < (long long)N * 32; j += stride) acc1[j] = 0.0f;
  for (long long j = i; j < (long long)N * 16; j += stride) acc2[j] = 0.0f;
}

// ---------------------------------------------------------------------------
// degree: deg[dst] += 1 over E edges (self-loops already counted in init)
// ---------------------------------------------------------------------------
__global__ void gcn_degree(const long long* __restrict__ dst,
                           float* __restrict__ deg, long long E) {
  long long i = (long long)blockIdx.x * blockDim.x + threadIdx.x;
  long long stride = (long long)gridDim.x * blockDim.x;
  for (long long e = i; e < E; e += stride) {
    atomicAdd(&deg[(int)dst[e]], 1.0f);
  }
}

// deg -> deg^{-1/2} in place
__global__ void gcn_dinv(float* __restrict__ deg, int N) {
  int i = blockIdx.x * blockDim.x + threadIdx.x;
  if (i < N) {
    float d = deg[i];
    deg[i] = (d > 0.0f) ? rsqrtf(d) : 0.0f;
  }
}

// ---------------------------------------------------------------------------
// dense1: h1[16n x 32] = x[16n x 3 (pad to K=4)] @ W1[3x32 (pad to 4x32)]
// one wave per 16-node tile; two V_WMMA_F32_16X16X4_F32 per tile.
// A layout (16x4 f32): lanes 0-15 hold M=lane, K=0..1; lanes 16-31 K=2..3.
// B layout (4x16 f32): lanes 0-15 hold N=lane, K=0..1; lanes 16-31 K=2..3.
// C/D layout: VGPR r: lanes 0-15 -> (M=r, N=lane); lanes 16-31 -> (M=r+8).
// Fast path (tile fully in range, wave-uniform test) has zero exec churn.
// ---------------------------------------------------------------------------
__global__ void gcn_dense1_wmma(const float* __restrict__ x,
                                const float* __restrict__ W1,
                                float* __restrict__ h1, int N) {
  int wave = (int)((blockIdx.x * blockDim.x + threadIdx.x) >> 5);
  int lane = threadIdx.x & 31;
  int base = wave * 16;
  if (base >= N) return;                 // wave-uniform
  const bool full = (base + 16 <= N);    // wave-uniform
  int half = lane >> 4;                  // 0: K=0,1   1: K=2,3
  int l = lane & 15;
  int row = base + l;

  v2f a;
  if (full) {
    if (half == 0) { a.x = x[row * 3 + 0]; a.y = x[row * 3 + 1]; }
    else           { a.x = x[row * 3 + 2]; a.y = 0.0f; }
  } else {
    bool rowOK = row < N;
    if (half == 0) {
      a.x = rowOK ? x[row * 3 + 0] : 0.0f;
      a.y = rowOK ? x[row * 3 + 1] : 0.0f;
    } else {
      a.x = rowOK ? x[row * 3 + 2] : 0.0f;
      a.y = 0.0f;
    }
  }

  #pragma unroll
  for (int t = 0; t < 2; ++t) {          // output columns 0-15, 16-31
    int col = t * 16 + l;
    v2f b;
    if (half == 0) { b.x = W1[0 * 32 + col]; b.y = W1[1 * 32 + col]; }
    else           { b.x = W1[2 * 32 + col]; b.y = 0.0f; }
    v8f c = {};
    c = __builtin_amdgcn_wmma_f32_16x16x4_f32(false, a, false, b,
                                              (short)0, c, false, false);
    if (full) {
      #pragma unroll
      for (int r = 0; r < 8; ++r) {
        int m = base + r + half * 8;
        h1[(long long)m * 32 + col] = c[r];
      }
    } else {
      #pragma unroll
      for (int r = 0; r < 8; ++r) {
        int m = base + r + half * 8;
        if (m < N) h1[(long long)m * 32 + col] = c[r];
      }
    }
  }
}

// ---------------------------------------------------------------------------
// dense2: h2[16n x 16] = a1[16n x 32] @ W2[32x16]; 8 chained K=4 WMMAs
// ---------------------------------------------------------------------------
__global__ void gcn_dense2_wmma(const float* __restrict__ a1,
                                const float* __restrict__ W2,
                                float* __restrict__ h2, int N) {
  int wave = (int)((blockIdx.x * blockDim.x + threadIdx.x) >> 5);
  int lane = threadIdx.x & 31;
  int base = wave * 16;
  if (base >= N) return;                 // wave-uniform
  const bool full = (base + 16 <= N);    // wave-uniform
  int half = lane >> 4;
  int l = lane & 15;
  int row = base + l;

  v8f c = {};
  if (full) {
    #pragma unroll
    for (int k = 0; k < 32; k += 4) {
      int k0 = k + half * 2;             // this lane-half's K pair
      v2f a, b;
      a.x = a1[(long long)row * 32 + k0];
      a.y = a1[(long long)row * 32 + k0 + 1];
      b.x = W2[(k0)     * 16 + l];
      b.y = W2[(k0 + 1) * 16 + l];
      c = __builtin_amdgcn_wmma_f32_16x16x4_f32(false, a, false, b,
                                                (short)0, c, false, false);
    }
    #pragma unroll
    for (int r = 0; r < 8; ++r) {
      int m = base + r + half * 8;
      h2[(long long)m * 16 + l] = c[r];
    }
  } else {
    bool rowOK = row < N;
    #pragma unroll
    for (int k = 0; k < 32; k += 4) {
      int k0 = k + half * 2;
      v2f a, b;
      a.x = rowOK ? a1[(long long)row * 32 + k0]     : 0.0f;
      a.y = rowOK ? a1[(long long)row * 32 + k0 + 1] : 0.0f;
      b.x = W2[(k0)     * 16 + l];
      b.y = W2[(k0 + 1) * 16 + l];
      c = __builtin_amdgcn_wmma_f32_16x16x4_f32(false, a, false, b,
                                                (short)0, c, false, false);
    }
    #pragma unroll
    for (int r = 0; r < 8; ++r) {
      int m = base + r + half * 8;
      if (m < N) h2[(long long)m * 16 + l] = c[r];
    }
  }
}

// ---------------------------------------------------------------------------
// scatter, 32 features: lane = feature, one edge per 32-lane group
// acc[dst*32+f] += h[src*32+f] * dinv[src]*dinv[dst]
// atomics per edge form one contiguous 128B burst -> coalesced L2 atomics
// ---------------------------------------------------------------------------
__global__ void gcn_scatter32(const long long* __restrict__ src,
                              const long long* __restrict__ dst,
                              const float* __restrict__ dinv,
                              const float* __restrict__ h,
                              float* __restrict__ acc, long long E) {
  long long idx = (long long)blockIdx.x * blockDim.x + threadIdx.x;
  long long stride = (long long)gridDim.x * blockDim.x;
  long long total = E * 32;
  for (; idx < total; idx += stride) {
    long long e = idx >> 5;
    int f = (int)(idx & 31);
    int s = (int)src[e];
    int d = (int)dst[e];
    float nrm = dinv[s] * dinv[d];
    atomicAdd(&acc[(long long)d * 32 + f], h[(long long)s * 32 + f] * nrm);
  }
}

// scatter, 16 features: 2 edges per 32-lane group (64B burst per edge)
__global__ void gcn_scatter16(const long long* __restrict__ src,
                              const long long* __restrict__ dst,
                              const float* __restrict__ dinv,
                              const float* __restrict__ h,
                              float* __restrict__ acc, long long E) {
  long long idx = (long long)blockIdx.x * blockDim.x + threadIdx.x;
  long long stride = (long long)gridDim.x * blockDim.x;
  long long total = E * 16;
  for (; idx < total; idx += stride) {
    long long e = idx >> 4;
    int f = (int)(idx & 15);
    int s = (int)src[e];
    int d = (int)dst[e];
    float nrm = dinv[s] * dinv[d];
    atomicAdd(&acc[(long long)d * 16 + f], h[(long long)s * 16 + f] * nrm);
  }
}

// ---------------------------------------------------------------------------
// a1 = relu(acc1 + h1*dinv^2 + b1)   (in place on acc1), 32 features
// ---------------------------------------------------------------------------
__global__ void gcn_relu_bias32(float* __restrict__ acc,
                                const float* __restrict__ h,
                                const float* __restrict__ dinv,
                                const float* __restrict__ b, int N) {
  long long idx = (long long)blockIdx.x * blockDim.x + threadIdx.x;
  if (idx >= (long long)N * 32) return;
  int n = (int)(idx >> 5);
  int f = (int)(idx & 31);
  float di = dinv[n];
  float v = acc[idx] + h[idx] * di * di + b[f];
  acc[idx] = v > 0.0f ? v : 0.0f;
}

// ---------------------------------------------------------------------------
// final: per node: v = relu(acc2 + h2*dinv^2 + b2); out = v . Wfc + bfc
// ---------------------------------------------------------------------------
__global__ void gcn_final(const float* __restrict__ acc2,
                          const float* __restrict__ h2,
                          const float* __restrict__ dinv,
                          const float* __restrict__ b2,
                          const float* __restrict__ Wfc,
                          const float* __restrict__ bfc,
                          float* __restrict__ out, int N) {
  int n = blockIdx.x * blockDim.x + threadIdx.x;
  if (n >= N) return;
  float di = dinv[n];
  float d2 = di * di;
  float s = 0.0f;
  #pragma unroll
  for (int f = 0; f < 16; ++f) {
    float v = acc2[(long long)n * 16 + f] + h2[(long long)n * 16 + f] * d2 + b2[f];
    v = v > 0.0f ? v : 0.0f;
    s += v * Wfc[f];
  }
  out[n] = s + bfc[0];
}

// ---------------------------------------------------------------------------
extern "C" void kernel_launch(void* const* d_in, const int* in_sizes, int n_in,
                              void* d_out, int out_size, void* d_ws, size_t ws_size,
                              hipStream_t stream) {
  const float*     x   = (const float*)d_in[0];
  const long long* ei  = (const long long*)d_in[1];
  const float*     W1  = (const float*)d_in[2];
  const float*     b1  = (const float*)d_in[3];
  const float*     W2  = (const float*)d_in[4];
  const float*     b2  = (const float*)d_in[5];
  const float*     Wfc = (const float*)d_in[6];
  const float*     bfc = (const float*)d_in[7];
  float* out = (float*)d_out;

  const int       N = in_sizes[0] / 3;
  const long long E = (long long)in_sizes[1] / 2;
  const long long* src = ei;       // edge_index[0, :]
  const long long* dst = ei + E;   // edge_index[1, :]

  // workspace layout (floats)
  float* ws   = (float*)d_ws;
  float* deg  = ws;                         // N      (becomes dinv)
  float* h1   = deg  + N;                   // N*32
  float* acc1 = h1   + (long long)N * 32;   // N*32   (becomes a1 after relu)
  float* h2   = acc1 + (long long)N * 32;   // N*16
  float* acc2 = h2   + (long long)N * 16;   // N*16

  const int B = 256;
  long long maxInit = (long long)N * 32;
  int gInit = (int)((maxInit + B - 1) / B);
  gcn_init<<<gInit, B, 0, stream>>>(deg, acc1, acc2, N);

  int gDeg = (int)((E + B - 1) / B);
  gcn_degree<<<gDeg, B, 0, stream>>>(dst, deg, E);

  gcn_dinv<<<(N + B - 1) / B, B, 0, stream>>>(deg, N);

  // dense layer 1: one wave per 16-node tile, 8 waves per block
  int tiles = (N + 15) / 16;
  int gD = (tiles + 7) / 8;
  gcn_dense1_wmma<<<gD, B, 0, stream>>>(x, W1, h1, N);

  long long w1 = E * 32;
  int gS1 = (int)((w1 + B - 1) / B);
  gcn_scatter32<<<gS1, B, 0, stream>>>(src, dst, deg, h1, acc1, E);

  int gR1 = (int)(((long long)N * 32 + B - 1) / B);
  gcn_relu_bias32<<<gR1, B, 0, stream>>>(acc1, h1, deg, b1, N);

  gcn_dense2_wmma<<<gD, B, 0, stream>>>(acc1, W2, h2, N);

  long long w2 = E * 16;
  int gS2 = (int)((w2 + B - 1) / B);
  gcn_scatter16<<<gS2, B, 0, stream>>>(src, dst, deg, h2, acc2, E);

  gcn_final<<<(N + B - 1) / B, B, 0, stream>>>(acc2, h2, deg, b2, Wfc, bfc, out, N);
}